// HeadTargetLayer_37598143710088
// MI455X (gfx1250) — compile-verified
//
#include <hip/hip_runtime.h>
#include <hip/hip_bf16.h>
#include <stdint.h>

#define UPPER_TH 0.4f
#define LOWER_TH 0.1f
#define BG_LABEL 20
#define MAXPOS 2048
#define LIST_CAP 16384
#define TPB 256

typedef float v2f __attribute__((ext_vector_type(2)));
typedef float v8f __attribute__((ext_vector_type(8)));

// ---------------- helpers ----------------

__device__ __forceinline__ float iou_pair(float p0, float p1, float p2, float p3,
                                          float g0, float g1, float g2, float g3) {
  float a1 = (p2 - p0 + 1.f) * (p3 - p1 + 1.f);
  float a2 = (g2 - g0 + 1.f) * (g3 - g1 + 1.f);
  float iw = fminf(p2, g2) - fmaxf(p0, g0) + 1.f; iw = fmaxf(iw, 0.f);
  float ih = fminf(p3, g3) - fmaxf(p1, g1) + 1.f; ih = fmaxf(ih, 0.f);
  float inter = iw * ih;
  return inter / (a1 + a2 - inter);
}

__device__ __forceinline__ float smooth_l1(float d) {
  float ad = fabsf(d);
  return (ad < 0.01f) ? (50.f * d * d) : (ad - 0.005f);
}

__device__ __forceinline__ unsigned rotl32(unsigned x, int r) {
  return (x << r) | (x >> (32 - r));
}

// Exact JAX threefry2x32 bits for jax.random.uniform(key(42), (L,)), element i.
__device__ unsigned threefry_bits(unsigned i, unsigned half) {
  unsigned k  = (i < half) ? i : (i - half);
  unsigned x0 = k, x1 = k + half;
  const unsigned ks0 = 0u, ks1 = 42u, ks2 = 0u ^ 42u ^ 0x1BD11BDAu;
  x0 += ks0; x1 += ks1;
#define TF_R(r) { x0 += x1; x1 = rotl32(x1, (r)); x1 ^= x0; }
  TF_R(13) TF_R(15) TF_R(26) TF_R(6)   x0 += ks1; x1 += ks2 + 1u;
  TF_R(17) TF_R(29) TF_R(16) TF_R(24)  x0 += ks2; x1 += ks0 + 2u;
  TF_R(13) TF_R(15) TF_R(26) TF_R(6)   x0 += ks0; x1 += ks1 + 3u;
  TF_R(17) TF_R(29) TF_R(16) TF_R(24)  x0 += ks1; x1 += ks2 + 4u;
  TF_R(13) TF_R(15) TF_R(26) TF_R(6)   x0 += ks2; x1 += ks0 + 5u;
#undef TF_R
  return (i < half) ? x0 : x1;
}

__device__ __forceinline__ unsigned long long ullmax2(unsigned long long a, unsigned long long b) {
  return a > b ? a : b;
}

// ---------------- kernels ----------------

// ctrl layout (ints): [0]=num_neg(atomic) [1]=listcount(atomic) [2]=mode(0=BIN,1=ALL,2=NONE)
// [3]=bin_sel [4]=r_sel [5]=num_pos [6]=bg_num [8..9]=threshold key (u64)
__global__ void k_init(int* ctrl, int* hist) {
  int t = threadIdx.x;
  if (t < 16) ctrl[t] = 0;
  hist[t] = 0;
}

// per-row: class argmax, logsumexp, pred box, threefry bits, gtassign init
__global__ void k_rowinit(const float* __restrict__ rois, const float* __restrict__ cls,
                          const float* __restrict__ deltas, int L, int C,
                          float* __restrict__ pred, float* __restrict__ lse,
                          unsigned* __restrict__ vbits, int* __restrict__ gtassign) {
  int i = blockIdx.x * TPB + threadIdx.x;
  if (i >= L) return;
  const float* s = cls + (size_t)i * C;
  float mx = s[0]; int am = 0;
  for (int j = 1; j < C; ++j) { float x = s[j]; if (x > mx) { mx = x; am = j; } }
  float se = 0.f;
  for (int j = 0; j < C; ++j) se += expf(s[j] - mx);
  lse[i] = mx + logf(se);
  const float* d = deltas + (size_t)i * 4 * C + (size_t)am * 4;
  float4 p;
  p.x = rois[(size_t)i * 5 + 1] + d[0];
  p.y = rois[(size_t)i * 5 + 2] + d[1];
  p.z = rois[(size_t)i * 5 + 3] + d[2];
  p.w = rois[(size_t)i * 5 + 4] + d[3];
  ((float4*)pred)[i] = p;
  vbits[i] = threefry_bits((unsigned)i, (unsigned)(L / 2)) >> 9;  // 23 random bits
  gtassign[i] = -1;
}

// per-row best IoU + first-occurrence argmax over M gts (gt boxes staged in LDS)
__global__ void k_rowbest(const float* __restrict__ pred, const float* __restrict__ gt,
                          int M, int L, float* __restrict__ bestp, int* __restrict__ idxp) {
  __shared__ float4 sgt[512];
  int Mc = M > 512 ? 512 : M;
  for (int j = threadIdx.x; j < Mc; j += TPB) sgt[j] = ((const float4*)gt)[j];
  __syncthreads();
  int i = blockIdx.x * TPB + threadIdx.x;
  if (i >= L) return;
  float4 p = ((const float4*)pred)[i];
  float best = -1.f; int bi = 0;
  for (int j = 0; j < Mc; ++j) {
    float4 g = sgt[j];
    float ov = iou_pair(p.x, p.y, p.z, p.w, g.x, g.y, g.z, g.w);
    if (ov > best) { best = ov; bi = j; }
  }
  bestp[i] = best; idxp[i] = bi;
}

// per-gt argmax over all rows; key = (ov_bits<<32)|~i so ties pick smallest i
__global__ void k_gtargmax(const float* __restrict__ pred, const float* __restrict__ gt,
                           int L, int* __restrict__ gtwin) {
  int j = blockIdx.x;
  float4 g = ((const float4*)gt)[j];
  unsigned long long best = 0ULL;
  for (int i = threadIdx.x; i < L; i += TPB) {
    float4 p = ((const float4*)pred)[i];
    float ov = iou_pair(p.x, p.y, p.z, p.w, g.x, g.y, g.z, g.w);
    unsigned long long key =
        ((unsigned long long)__float_as_uint(ov) << 32) | (unsigned)(~(unsigned)i);
    best = ullmax2(best, key);
  }
  __shared__ unsigned long long sb[TPB];
  sb[threadIdx.x] = best; __syncthreads();
  for (int s = TPB / 2; s > 0; s >>= 1) {
    if (threadIdx.x < s) sb[threadIdx.x] = ullmax2(sb[threadIdx.x], sb[threadIdx.x + s]);
    __syncthreads();
  }
  if (threadIdx.x == 0) gtwin[j] = (int)(~(unsigned)(sb[0] & 0xFFFFFFFFull));
}

// scatter gt index onto its winner row (duplicates -> max j, deterministic)
__global__ void k_gtscatter(const int* __restrict__ gtwin, int M, int* __restrict__ gtassign) {
  int j = blockIdx.x * TPB + threadIdx.x;
  if (j < M) atomicMax(&gtassign[gtwin[j]], j);
}

// finalize matches, pos flags, neg count, neg-u histogram (top 8 of 23 bits)
__global__ void k_finalize(const float* __restrict__ bestp, const int* __restrict__ idxp,
                           const int* __restrict__ gtassign, const unsigned* __restrict__ vbits,
                           int L, int* __restrict__ matches, int* __restrict__ posflag,
                           int* ctrl, int* hist) {
  int i = blockIdx.x * TPB + threadIdx.x;
  if (i >= L) return;
  float bp = bestp[i];
  int m = (bp >= UPPER_TH) ? idxp[i] : -1;
  int ga = gtassign[i];
  if (ga >= 0) m = ga;
  if (bp < LOWER_TH) m = -2;
  matches[i] = m;
  posflag[i] = (m >= 0) ? 1 : 0;
  if (m == -2) {
    atomicAdd(&ctrl[0], 1);
    atomicAdd(&hist[vbits[i] >> 15], 1);
  }
}

// per-block inclusive scan of pos flags
__global__ void k_scan(const int* __restrict__ posflag, int L,
                       int* __restrict__ posscan, int* __restrict__ blocksum) {
  __shared__ int sh[TPB];
  int i = blockIdx.x * TPB + threadIdx.x;
  sh[threadIdx.x] = (i < L) ? posflag[i] : 0;
  __syncthreads();
  for (int off = 1; off < TPB; off <<= 1) {
    int v = ((int)threadIdx.x >= off) ? sh[threadIdx.x - off] : 0;
    __syncthreads();
    sh[threadIdx.x] += v;
    __syncthreads();
  }
  if (i < L) posscan[i] = sh[threadIdx.x];
  if (threadIdx.x == TPB - 1) blocksum[blockIdx.x] = sh[TPB - 1];
}

// serial control: block offsets, num_pos, bg_num, radix-bin boundary for neg sampling
__global__ void k_control(const int* __restrict__ blocksum, int nblk,
                          const int* __restrict__ hist, int* __restrict__ blockoff, int* ctrl) {
  if (threadIdx.x != 0) return;
  int run = 0;
  for (int b = 0; b < nblk; ++b) { blockoff[b] = run; run += blocksum[b]; }
  int num_pos = run;
  int bg = (int)rintf((float)num_pos * 1.0f);
  int num_neg = ctrl[0];
  int mode = 0, bin = 0, r = 0;
  if (bg <= 0) mode = 2;
  else if (bg >= num_neg) mode = 1;
  else {
    int cum = 0;
    for (int b = 0; b < 256; ++b) {
      int h = hist[b];
      if (cum + h >= bg) { bin = b; r = bg - cum; break; }
      cum += h;
    }
  }
  ctrl[2] = mode; ctrl[3] = bin; ctrl[4] = r; ctrl[5] = num_pos; ctrl[6] = bg;
}

// collect composite keys of negatives landing in the boundary bin
__global__ void k_collect(const int* __restrict__ matches, const unsigned* __restrict__ vbits,
                          int L, int* ctrl, unsigned long long* __restrict__ list) {
  int i = blockIdx.x * TPB + threadIdx.x;
  if (i >= L) return;
  if (ctrl[2] != 0) return;
  if (matches[i] != -2) return;
  unsigned vv = vbits[i];
  if ((int)(vv >> 15) != ctrl[3]) return;
  int p = atomicAdd(&ctrl[1], 1);
  if (p < LIST_CAP) list[p] = ((unsigned long long)vv << 32) | (unsigned)i;
}

// exact rank-select of the global bg_num-th smallest key inside the boundary bin
__global__ void k_thresh(int* ctrl, const unsigned long long* __restrict__ list) {
  unsigned long long* T = (unsigned long long*)(ctrl + 8);
  int mode = ctrl[2];
  if (mode == 1) { if (threadIdx.x == 0) *T = ~0ULL; return; }
  if (mode == 2) { if (threadIdx.x == 0) *T = 0ULL;  return; }
  int n = ctrl[1]; if (n > LIST_CAP) n = LIST_CAP;
  int r = ctrl[4];
  for (int idx = threadIdx.x; idx < n; idx += blockDim.x) {
    unsigned long long key = list[idx];
    int rank = 0;
    for (int j = 0; j < n; ++j) rank += (list[j] < key) ? 1 : 0;
    if (rank == r - 1) *T = key;  // unique winner (keys carry the row index)
  }
}

// weighted CE partial sums (deterministic LDS tree per block)
__global__ void k_cls(const float* __restrict__ cls, const float* __restrict__ lse,
                      const int* __restrict__ matches, const unsigned* __restrict__ vbits,
                      const int* __restrict__ gtcls, int L, int C,
                      const int* __restrict__ ctrl, float* __restrict__ clspart) {
  int i = blockIdx.x * TPB + threadIdx.x;
  float wce = 0.f, w = 0.f;
  if (i < L) {
    int m = matches[i];
    bool pos = (m >= 0);
    bool sel = pos;
    if (!pos && m == -2) {
      int mode = ctrl[2];
      if (mode == 1) sel = true;
      else if (mode == 0) {
        unsigned long long T = *(const unsigned long long*)(ctrl + 8);
        unsigned long long key = ((unsigned long long)vbits[i] << 32) | (unsigned)i;
        sel = (key <= T);
      }
    }
    if (sel) {
      int label = pos ? gtcls[m] : BG_LABEL;
      w = 1.f;
      wce = lse[i] - cls[(size_t)i * C + label];
    }
  }
  __shared__ float sa[TPB], sw[TPB];
  sa[threadIdx.x] = wce; sw[threadIdx.x] = w; __syncthreads();
  for (int s = TPB / 2; s > 0; s >>= 1) {
    if (threadIdx.x < s) { sa[threadIdx.x] += sa[threadIdx.x + s]; sw[threadIdx.x] += sw[threadIdx.x + s]; }
    __syncthreads();
  }
  if (threadIdx.x == 0) { clspart[blockIdx.x * 2] = sa[0]; clspart[blockIdx.x * 2 + 1] = sw[0]; }
}

// pos_idx compaction: pos rows ascending then non-pos ascending; gather pred/gt/valid
__global__ void k_build(const float* __restrict__ pred, const float* __restrict__ gt,
                        const int* __restrict__ matches, const int* __restrict__ posflag,
                        const int* __restrict__ posscan, const int* __restrict__ blockoff,
                        const int* __restrict__ ctrl, int L,
                        float* __restrict__ predp, float* __restrict__ gtp,
                        float* __restrict__ validv) {
  int i = blockIdx.x * TPB + threadIdx.x;
  if (i >= L) return;
  int incl = posscan[i] + blockoff[blockIdx.x];
  int excl = incl - posflag[i];
  int num_pos = ctrl[5];
  int oidx = posflag[i] ? excl : (num_pos + (i - excl));
  if (oidx < MAXPOS) {
    ((float4*)predp)[oidx] = ((const float4*)pred)[i];
    int m = matches[i]; if (m < 0) m = 0;
    ((float4*)gtp)[oidx] = ((const float4*)gt)[m];
    validv[oidx] = posflag[i] ? 1.f : 0.f;
  }
}

// pairwise smooth-L1 over 2048x2048 pairs.
// One 16x16 (a,b) tile per wave; pair_w tile = valid_a (X) valid_b computed with
// V_WMMA_F32_16X16X4_F32 rank-1 outer product (A: K=0 column, B: K=0 row).
// D layout: VGPR v, lane l -> m = v + 8*(l>=16), n = l&15 (matches pair assignment).
__global__ void k_pairwise(const float* __restrict__ predp, const float* __restrict__ gtp,
                           const float* __restrict__ validv, float* __restrict__ bbpart) {
  int lane = threadIdx.x & 31;
  int wave = threadIdx.x >> 5;
  int tile = blockIdx.x * 8 + wave;     // 16384 tiles total, always valid
  int ta = tile >> 7;                   // a-tile (gt rows), 0..127
  int tb = tile & 127;                  // b-tile (pred rows), 0..127
  int n  = lane & 15;
  int hi = lane >> 4;

  v2f A = {}; v2f B = {};
  A[0] = (lane < 16) ? validv[ta * 16 + lane] : 0.f;  // A[m, K=0] = valid_a[m]
  B[0] = (lane < 16) ? validv[tb * 16 + lane] : 0.f;  // B[K=0, n] = valid_b[n]
  v8f Cz = {};
  v8f D = __builtin_amdgcn_wmma_f32_16x16x4_f32(false, A, false, B,
                                                (short)0, Cz, false, false);

  float4 pb = ((const float4*)predp)[tb * 16 + n];
  float acc = 0.f;
#pragma unroll
  for (int v = 0; v < 8; ++v) {
    int m = hi * 8 + v;
    float4 ga = ((const float4*)gtp)[ta * 16 + m];
    float s = smooth_l1(pb.x - ga.x) + smooth_l1(pb.y - ga.y) +
              smooth_l1(pb.z - ga.z) + smooth_l1(pb.w - ga.w);
    acc += D[v] * s;
  }
  for (int off = 16; off > 0; off >>= 1) acc += __shfl_xor(acc, off, 32);
  __shared__ float sh[8];
  if (lane == 0) sh[wave] = acc;
  __syncthreads();
  if (threadIdx.x == 0) {
    float t = 0.f;
    for (int wv = 0; wv < 8; ++wv) t += sh[wv];
    bbpart[blockIdx.x] = t;
  }
}

// final deterministic reduction -> d_out[0]=cls_loss, d_out[1]=bbox_loss
__global__ void k_final(const float* __restrict__ clspart, int nblk,
                        const float* __restrict__ bbpart, int nbb, float* __restrict__ out) {
  __shared__ float sh[TPB];
  float t = 0.f;
  for (int idx = threadIdx.x; idx < nbb; idx += TPB) t += bbpart[idx];
  sh[threadIdx.x] = t; __syncthreads();
  for (int s = TPB / 2; s > 0; s >>= 1) {
    if (threadIdx.x < s) sh[threadIdx.x] += sh[threadIdx.x + s];
    __syncthreads();
  }
  if (threadIdx.x == 0) {
    float a = 0.f, w = 0.f;
    for (int b = 0; b < nblk; ++b) { a += clspart[2 * b]; w += clspart[2 * b + 1]; }
    out[0] = a / w;
    out[1] = sh[0];
  }
}

// ---------------- host ----------------

extern "C" void kernel_launch(void* const* d_in, const int* in_sizes, int n_in,
                              void* d_out, int out_size, void* d_ws, size_t ws_size,
                              hipStream_t stream) {
  const float* rois   = (const float*)d_in[0];
  const float* cls    = (const float*)d_in[1];
  const float* deltas = (const float*)d_in[2];
  const float* gt     = (const float*)d_in[3];
  const int*   gtcls  = (const int*)d_in[4];
  float* out = (float*)d_out;

  int L = in_sizes[0] / 5;
  int C = in_sizes[1] / L;
  int M = in_sizes[4];
  int rowBlk = (L + TPB - 1) / TPB;

  // workspace carve-up (256B aligned)
  char* base = (char*)d_ws;
  size_t off = 0;
  auto carve = [&](size_t bytes) -> char* {
    char* p = base + off;
    off = (off + bytes + 255) & ~(size_t)255;
    return p;
  };
  float*    pred     = (float*)carve((size_t)L * 4 * sizeof(float));
  float*    lse      = (float*)carve((size_t)L * sizeof(float));
  float*    bestp    = (float*)carve((size_t)L * sizeof(float));
  int*      idxp     = (int*)carve((size_t)L * sizeof(int));
  unsigned* vbits    = (unsigned*)carve((size_t)L * sizeof(unsigned));
  int*      matches  = (int*)carve((size_t)L * sizeof(int));
  int*      gtassign = (int*)carve((size_t)L * sizeof(int));
  int*      posflag  = (int*)carve((size_t)L * sizeof(int));
  int*      posscan  = (int*)carve((size_t)L * sizeof(int));
  int*      blocksum = (int*)carve((size_t)rowBlk * sizeof(int));
  int*      blockoff = (int*)carve((size_t)rowBlk * sizeof(int));
  int*      gtwin    = (int*)carve((size_t)M * sizeof(int));
  int*      hist     = (int*)carve(256 * sizeof(int));
  int*      ctrl     = (int*)carve(64);
  unsigned long long* list = (unsigned long long*)carve(LIST_CAP * sizeof(unsigned long long));
  float*    predp    = (float*)carve((size_t)MAXPOS * 4 * sizeof(float));
  float*    gtp      = (float*)carve((size_t)MAXPOS * 4 * sizeof(float));
  float*    validv   = (float*)carve((size_t)MAXPOS * sizeof(float));
  float*    clspart  = (float*)carve((size_t)rowBlk * 2 * sizeof(float));
  const int BBBLK = (MAXPOS / 16) * (MAXPOS / 16) / 8;  // 2048 blocks, 8 tiles each
  float*    bbpart   = (float*)carve((size_t)BBBLK * sizeof(float));
  (void)ws_size; (void)n_in; (void)out_size;

  k_init<<<1, TPB, 0, stream>>>(ctrl, hist);
  k_rowinit<<<rowBlk, TPB, 0, stream>>>(rois, cls, deltas, L, C, pred, lse, vbits, gtassign);
  k_rowbest<<<rowBlk, TPB, 0, stream>>>(pred, gt, M, L, bestp, idxp);
  k_gtargmax<<<M, TPB, 0, stream>>>(pred, gt, L, gtwin);
  k_gtscatter<<<(M + TPB - 1) / TPB, TPB, 0, stream>>>(gtwin, M, gtassign);
  k_finalize<<<rowBlk, TPB, 0, stream>>>(bestp, idxp, gtassign, vbits, L, matches, posflag, ctrl, hist);
  k_scan<<<rowBlk, TPB, 0, stream>>>(posflag, L, posscan, blocksum);
  k_control<<<1, 1, 0, stream>>>(blocksum, rowBlk, hist, blockoff, ctrl);
  k_collect<<<rowBlk, TPB, 0, stream>>>(matches, vbits, L, ctrl, list);
  k_thresh<<<1, TPB, 0, stream>>>(ctrl, list);
  k_cls<<<rowBlk, TPB, 0, stream>>>(cls, lse, matches, vbits, gtcls, L, C, ctrl, clspart);
  k_build<<<rowBlk, TPB, 0, stream>>>(pred, gt, matches, posflag, posscan, blockoff, ctrl, L,
                                      predp, gtp, validv);
  k_pairwise<<<BBBLK, TPB, 0, stream>>>(predp, gtp, validv, bbpart);
  k_final<<<1, TPB, 0, stream>>>(clspart, rowBlk, bbpart, BBBLK, out);
}